// ContextQueryAttention_72791105733170
// MI455X (gfx1250) — compile-verified
//
#include <hip/hip_runtime.h>

static constexpr int   B   = 64;
static constexpr int   LC  = 1024;
static constexpr int   LQ  = 128;
static constexpr int   D   = 128;
static constexpr float NEG_BIG = 1.0e12f;

typedef __attribute__((ext_vector_type(2))) float v2f;
typedef __attribute__((ext_vector_type(8))) float v8f;

__device__ __forceinline__ v8f wmma4(v2f a, v2f b, v8f c) {
  // 8 args: (neg_a, A, neg_b, B, c_mod, C, reuse_a, reuse_b)
  return __builtin_amdgcn_wmma_f32_16x16x4_f32(false, a, false, b, (short)0, c,
                                               false, false);
}

// ---------------------------------------------------------------------------
// K1: S[b,c,q] = x_cont@W0 + x_ques@W1 + (x_cont*W2) . x_ques   (into S buf)
// grid (LC/16, B), block 128 (4 waves, each wave: 2 q-tiles of 16)
// ---------------------------------------------------------------------------
__global__ __launch_bounds__(128)
void k1_scores(const float* __restrict__ xc, const float* __restrict__ xq,
               const float* __restrict__ W0, const float* __restrict__ W1,
               const float* __restrict__ W2, float* __restrict__ S)
{
  const int b  = blockIdx.y;
  const int c0 = blockIdx.x * 16;
  const int t  = threadIdx.x;

  __shared__ float s_ques[LQ];
  __shared__ float s_cont[16];

  // Bias terms: one dot(D) per thread
  {
    const float* xr = xq + ((size_t)b * LQ + t) * D;
    float acc = 0.f;
    #pragma unroll 8
    for (int d = 0; d < D; ++d) acc += xr[d] * W1[d];
    s_ques[t] = acc;
    if (t < 16) {
      const float* cr = xc + ((size_t)b * LC + c0 + t) * D;
      float a2 = 0.f;
      #pragma unroll 8
      for (int d = 0; d < D; ++d) a2 += cr[d] * W0[d];
      s_cont[t] = a2;
    }
  }
  __syncthreads();

  const int lane = t & 31;
  const int wave = t >> 5;
  const int ml   = lane & 15;
  const int hl   = lane >> 4;
  const int q0a  = (wave * 2) * 16;
  const int q0b  = q0a + 16;

  const float* arow  = xc + ((size_t)b * LC + c0 + ml) * D;
  const float* brow0 = xq + ((size_t)b * LQ + q0a + ml) * D;
  const float* brow1 = xq + ((size_t)b * LQ + q0b + ml) * D;

  v8f acc0 = {}; v8f acc1 = {};
  #pragma unroll 4
  for (int k = 0; k < D; k += 4) {
    const int k2 = k + 2 * hl;
    v2f a;   a.x   = arow[k2] * W2[k2];  a.y   = arow[k2 + 1] * W2[k2 + 1];
    v2f bb0; bb0.x = brow0[k2];          bb0.y = brow0[k2 + 1];
    v2f bb1; bb1.x = brow1[k2];          bb1.y = brow1[k2 + 1];
    acc0 = wmma4(a, bb0, acc0);
    acc1 = wmma4(a, bb1, acc1);
  }

  #pragma unroll
  for (int r = 0; r < 8; ++r) {
    const int M = r + 8 * hl;
    const size_t base = ((size_t)b * LC + c0 + M) * LQ;
    S[base + q0a + ml] = acc0[r] + s_cont[M] + s_ques[q0a + ml];
    S[base + q0b + ml] = acc1[r] + s_cont[M] + s_ques[q0b + ml];
  }
}

// ---------------------------------------------------------------------------
// K2: in-place row softmax over q with query-length mask. One wave per row.
// grid (B*LC/8), block 256
// ---------------------------------------------------------------------------
__global__ __launch_bounds__(256)
void k2_row_softmax(float* __restrict__ S, const int* __restrict__ ques_len)
{
  const int wave = threadIdx.x >> 5;
  const int lane = threadIdx.x & 31;
  const int row  = blockIdx.x * 8 + wave;   // in [0, B*LC)
  const int b    = row >> 10;               // row / LC
  const int qlen = ques_len[b];
  float* rp = S + (size_t)row * LQ;

  float v[4];
  float m = -__builtin_inff();
  #pragma unroll
  for (int j = 0; j < 4; ++j) {
    const int q = lane + 32 * j;
    float x = rp[q];
    x -= (q < qlen) ? 0.f : NEG_BIG;
    v[j] = x;
    m = fmaxf(m, x);
  }
  #pragma unroll
  for (int off = 16; off > 0; off >>= 1) m = fmaxf(m, __shfl_xor(m, off, 32));
  float s = 0.f;
  #pragma unroll
  for (int j = 0; j < 4; ++j) { v[j] = __expf(v[j] - m); s += v[j]; }
  #pragma unroll
  for (int off = 16; off > 0; off >>= 1) s += __shfl_xor(s, off, 32);
  const float inv = 1.f / s;
  #pragma unroll
  for (int j = 0; j < 4; ++j) rp[lane + 32 * j] = v[j] * inv;
}

// ---------------------------------------------------------------------------
// K3a: column (over c) max & sum per (b,q), context-length mask.
// grid (LQ/32, B), block 256 = (32 q-lanes x 8 c-groups)
// ---------------------------------------------------------------------------
__global__ __launch_bounds__(256)
void k3a_col_stats(const float* __restrict__ S, const int* __restrict__ cont_len,
                   float* __restrict__ colmax, float* __restrict__ colsum)
{
  const int b  = blockIdx.y;
  const int q0 = blockIdx.x * 32;
  const int tq = threadIdx.x & 31;
  const int tg = threadIdx.x >> 5;
  const int clen = cont_len[b];
  const int q = q0 + tq;
  const float* sp = S + (size_t)b * LC * LQ;

  __shared__ float red[8][32];

  float m = -__builtin_inff();
  for (int c = tg; c < LC; c += 8) {
    float x = sp[(size_t)c * LQ + q];
    x -= (c < clen) ? 0.f : NEG_BIG;
    m = fmaxf(m, x);
  }
  red[tg][tq] = m;
  __syncthreads();
  if (tg == 0) {
    #pragma unroll
    for (int g = 1; g < 8; ++g) m = fmaxf(m, red[g][tq]);
    red[0][tq] = m;
  }
  __syncthreads();
  m = red[0][tq];
  __syncthreads();

  float s = 0.f;
  for (int c = tg; c < LC; c += 8) {
    float x = sp[(size_t)c * LQ + q];
    x -= (c < clen) ? 0.f : NEG_BIG;
    s += __expf(x - m);
  }
  red[tg][tq] = s;
  __syncthreads();
  if (tg == 0) {
    #pragma unroll
    for (int g = 1; g < 8; ++g) s += red[g][tq];
    colmax[b * LQ + q] = m;
    colsum[b * LQ + q] = s;
  }
}

// ---------------------------------------------------------------------------
// K3b: column softmax + transpose via 32x32 LDS tile -> S_T (B,LQ,LC)
// grid (LQ/32, B), block 256
// ---------------------------------------------------------------------------
__global__ __launch_bounds__(256)
void k3b_col_softmax_T(const float* __restrict__ S, const int* __restrict__ cont_len,
                       const float* __restrict__ colmax, const float* __restrict__ colsum,
                       float* __restrict__ ST)
{
  const int b  = blockIdx.y;
  const int q0 = blockIdx.x * 32;
  const int t  = threadIdx.x;
  const int clen = cont_len[b];
  __shared__ float tile[32][33];
  const float* sp = S  + (size_t)b * LC * LQ;
  float*       tp = ST + (size_t)b * LQ * LC;

  for (int cc = 0; cc < LC / 32; ++cc) {
    const int c0 = cc * 32;
    #pragma unroll
    for (int i = 0; i < 4; ++i) {
      const int idx = t + 256 * i;
      const int ci = idx >> 5, qi = idx & 31;
      const int c = c0 + ci,   q = q0 + qi;
      float x = sp[(size_t)c * LQ + q];
      x -= (c < clen) ? 0.f : NEG_BIG;
      tile[ci][qi] = __expf(x - colmax[b * LQ + q]) / colsum[b * LQ + q];
    }
    __syncthreads();
    #pragma unroll
    for (int i = 0; i < 4; ++i) {
      const int idx = t + 256 * i;
      const int qi = idx >> 5, ci = idx & 31;
      tp[(size_t)(q0 + qi) * LC + c0 + ci] = tile[ci][qi];
    }
    __syncthreads();
  }
}

// ---------------------------------------------------------------------------
// K4: Aint[b] = S_T[b] (128x1024) @ x_cont[b] (1024x128), WMMA K-loop = 1024
// grid (LQ/16, B), block 128 (4 waves x 2 d-tiles)
// ---------------------------------------------------------------------------
__global__ __launch_bounds__(128)
void k4_qc(const float* __restrict__ ST, const float* __restrict__ xc,
           float* __restrict__ Aint)
{
  const int b  = blockIdx.y;
  const int q0 = blockIdx.x * 16;
  const int lane = threadIdx.x & 31;
  const int wave = threadIdx.x >> 5;
  const int ml = lane & 15, hl = lane >> 4;
  const int d0a = (wave * 2) * 16, d0b = d0a + 16;

  const float* arow = ST + ((size_t)b * LQ + q0 + ml) * LC;
  const float* xcb  = xc + (size_t)b * LC * D;

  v8f acc0 = {}; v8f acc1 = {};
  for (int k = 0; k < LC; k += 4) {
    const int k2 = k + 2 * hl;
    v2f a;  a.x  = arow[k2];                       a.y  = arow[k2 + 1];
    v2f b0; b0.x = xcb[(size_t)k2 * D + d0a + ml]; b0.y = xcb[(size_t)(k2 + 1) * D + d0a + ml];
    v2f b1; b1.x = xcb[(size_t)k2 * D + d0b + ml]; b1.y = xcb[(size_t)(k2 + 1) * D + d0b + ml];
    acc0 = wmma4(a, b0, acc0);
    acc1 = wmma4(a, b1, acc1);
  }
  #pragma unroll
  for (int r = 0; r < 8; ++r) {
    const int M = r + 8 * hl;
    const size_t base = ((size_t)b * LQ + q0 + M) * D;
    Aint[base + d0a + ml] = acc0[r];
    Aint[base + d0b + ml] = acc1[r];
  }
}

// ---------------------------------------------------------------------------
// K5: c2q = S_bar @ x_ques ; q2c = S_bar @ Aint ; concat epilogue
// grid (LC/16, B), block 128 (4 waves x 2 d-tiles, shared A operand)
// ---------------------------------------------------------------------------
__global__ __launch_bounds__(128)
void k5_final(const float* __restrict__ Sbar, const float* __restrict__ xq,
              const float* __restrict__ Aint, const float* __restrict__ xc,
              float* __restrict__ out)
{
  const int b  = blockIdx.y;
  const int c0 = blockIdx.x * 16;
  const int lane = threadIdx.x & 31;
  const int wave = threadIdx.x >> 5;
  const int ml = lane & 15, hl = lane >> 4;
  const int d0a = (wave * 2) * 16, d0b = d0a + 16;

  const float* arow = Sbar + ((size_t)b * LC + c0 + ml) * LQ;
  const float* xqb  = xq   + (size_t)b * LQ * D;
  const float* aib  = Aint + (size_t)b * LQ * D;

  v8f c2q0 = {}, c2q1 = {}, q2c0 = {}, q2c1 = {};
  #pragma unroll 2
  for (int k = 0; k < LQ; k += 4) {
    const int k2 = k + 2 * hl;
    v2f a;   a.x   = arow[k2];                       a.y   = arow[k2 + 1];
    v2f bq0; bq0.x = xqb[(size_t)k2 * D + d0a + ml]; bq0.y = xqb[(size_t)(k2 + 1) * D + d0a + ml];
    v2f bq1; bq1.x = xqb[(size_t)k2 * D + d0b + ml]; bq1.y = xqb[(size_t)(k2 + 1) * D + d0b + ml];
    v2f ba0; ba0.x = aib[(size_t)k2 * D + d0a + ml]; ba0.y = aib[(size_t)(k2 + 1) * D + d0a + ml];
    v2f ba1; ba1.x = aib[(size_t)k2 * D + d0b + ml]; ba1.y = aib[(size_t)(k2 + 1) * D + d0b + ml];
    c2q0 = wmma4(a, bq0, c2q0);
    c2q1 = wmma4(a, bq1, c2q1);
    q2c0 = wmma4(a, ba0, q2c0);
    q2c1 = wmma4(a, ba1, q2c1);
  }

  #pragma unroll
  for (int r = 0; r < 8; ++r) {
    const int M = r + 8 * hl;
    const int row = c0 + M;
    const float* xrow = xc  + ((size_t)b * LC + row) * D;
    float*       orow = out + ((size_t)b * LC + row) * (4 * D);
    {
      const int d = d0a + ml;
      const float xv = xrow[d], cv = c2q0[r], qv = q2c0[r];
      orow[d] = xv; orow[D + d] = cv; orow[2 * D + d] = xv * cv; orow[3 * D + d] = xv * qv;
    }
    {
      const int d = d0b + ml;
      const float xv = xrow[d], cv = c2q1[r], qv = q2c1[r];
      orow[d] = xv; orow[D + d] = cv; orow[2 * D + d] = xv * cv; orow[3 * D + d] = xv * qv;
    }
  }
}

// ---------------------------------------------------------------------------
extern "C" void kernel_launch(void* const* d_in, const int* in_sizes, int n_in,
                              void* d_out, int out_size, void* d_ws, size_t ws_size,
                              hipStream_t stream)
{
  (void)in_sizes; (void)n_in; (void)out_size; (void)ws_size;

  const float* xc = (const float*)d_in[0];
  const float* xq = (const float*)d_in[1];
  const float* W0 = (const float*)d_in[2];
  const float* W1 = (const float*)d_in[3];
  const float* W2 = (const float*)d_in[4];
  const int* cont_len = (const int*)d_in[5];
  const int* ques_len = (const int*)d_in[6];

  float* out      = (float*)d_out;
  float* out_res  = out;                                   // (B,LC,4D)
  float* out_Sbar = out + (size_t)B * LC * 4 * D;          // (B,LC,LQ)
  float* out_ST   = out_Sbar + (size_t)B * LC * LQ;        // (B,LQ,LC)

  float* colmax = (float*)d_ws;                            // B*LQ
  float* colsum = colmax + (size_t)B * LQ;                 // B*LQ
  float* Aint   = colsum + (size_t)B * LQ;                 // B*LQ*D (4 MB)

  // Raw S is staged in the S_bar output region (same shape), then:
  //  - column stats + transposed column softmax -> S_T
  //  - in-place row softmax -> S_bar
  k1_scores       <<<dim3(LC / 16, B), 128, 0, stream>>>(xc, xq, W0, W1, W2, out_Sbar);
  k3a_col_stats   <<<dim3(LQ / 32, B), 256, 0, stream>>>(out_Sbar, cont_len, colmax, colsum);
  k3b_col_softmax_T<<<dim3(LQ / 32, B), 256, 0, stream>>>(out_Sbar, cont_len, colmax, colsum, out_ST);
  k2_row_softmax  <<<(B * LC) / 8, 256, 0, stream>>>(out_Sbar, ques_len);
  k4_qc           <<<dim3(LQ / 16, B), 128, 0, stream>>>(out_ST, xc, Aint);
  k5_final        <<<dim3(LC / 16, B), 128, 0, stream>>>(out_Sbar, xq, Aint, xc, out_res);
}